// VLinePooling2_21509196218384
// MI455X (gfx1250) — compile-verified
//
#include <hip/hip_runtime.h>

// Problem constants (from the reference)
#define B_  8
#define C_  32
#define H_  512
#define W_  512
#define L_  512
#define HW_ (H_ * W_)

constexpr int THREADS          = 256;  // 8 waves on wave32
constexpr int CG               = 16;   // channels per block -> 16*512*4 = 32 KB LDS accum
constexpr int TILE             = 1024; // pixels per tile (one float4 per thread)
constexpr int TILES_PER_BLOCK  = 8;    // 8192 pixels per block
// pixel chunks per (b, cg): HW / (TILE*TILES_PER_BLOCK) = 32

typedef __attribute__((ext_vector_type(4))) float v4f;
typedef __attribute__((ext_vector_type(4))) int   v4i;

// Relaxed, correctly-scoped FP atomics -> ds_add_f32 / global_atomic_add_f32
__device__ __forceinline__ void lds_fadd(float* p, float v) {
  __hip_atomic_fetch_add(p, v, __ATOMIC_RELAXED, __HIP_MEMORY_SCOPE_WORKGROUP);
}
__device__ __forceinline__ void glb_fadd(float* p, float v) {
  __hip_atomic_fetch_add(p, v, __ATOMIC_RELAXED, __HIP_MEMORY_SCOPE_AGENT);
}

// ---- gfx1250 async global->LDS path (guarded; sync fallback otherwise) ----
#if defined(__has_builtin)
#if __has_builtin(__builtin_amdgcn_global_load_async_to_lds_b128) && \
    __has_builtin(__builtin_amdgcn_s_wait_asynccnt)
#define USE_ASYNC 1
#endif
#endif
#ifndef USE_ASYNC
#define USE_ASYNC 0
#endif

#if USE_ASYNC
typedef __attribute__((address_space(1))) v4i* v4i_global_ptr;
typedef __attribute__((address_space(3))) v4i* v4i_lds_ptr;

__device__ __forceinline__ void async_copy_b128(const int* gsrc, int* ldst) {
  __builtin_amdgcn_global_load_async_to_lds_b128(
      (v4i_global_ptr)gsrc,
      (v4i_lds_ptr)ldst,
      /*offset=*/0, /*cpol=*/0);
}
#endif

__global__ __launch_bounds__(THREADS) void vlp_zero_kernel(float* __restrict__ p, int n) {
  int i = blockIdx.x * THREADS + threadIdx.x;
  if (i < n) p[i] = 0.0f;
}

__global__ __launch_bounds__(THREADS) void vlp_pool_kernel(
    const float* __restrict__ input,   // [B,C,H,W]
    const int*   __restrict__ indmap,  // [B,H,W]
    const int*   __restrict__ valid,   // [B,H,W]
    float*       __restrict__ out)     // [B,C,L], pre-zeroed
{
  __shared__ float acc[CG * L_];       // 32 KB per-block bin accumulators
  __shared__ int   lidx[2][TILE];      // 8 KB double-buffered packed indices
#if USE_ASYNC
  __shared__ int   lval[2][TILE];      // 8 KB raw valid staging
#endif

  const int tid    = threadIdx.x;
  const int b      = blockIdx.y;
  const int cg     = blockIdx.z;
  const int p_base = blockIdx.x * (TILES_PER_BLOCK * TILE);

  for (int j = tid; j < CG * L_; j += THREADS) acc[j] = 0.0f;

  const int*   imap    = indmap + (size_t)b * HW_;
  const int*   vmap    = valid  + (size_t)b * HW_;
  const float* in_base = input  + ((size_t)b * C_ + (size_t)cg * CG) * (size_t)HW_;

  int buf = 0;
#if USE_ASYNC
  // Prefetch tile 0 maps straight into LDS (bypasses VGPRs, ASYNCcnt-tracked).
  async_copy_b128(imap + p_base + tid * 4, &lidx[0][tid * 4]);
  async_copy_b128(vmap + p_base + tid * 4, &lval[0][tid * 4]);
#endif

  for (int t = 0; t < TILES_PER_BLOCK; ++t) {
    const int p0 = p_base + t * TILE;

#if USE_ASYNC
    __builtin_amdgcn_s_wait_asynccnt(0);
    __syncthreads();                           // tile t fully in LDS for all waves
    {
      int4 bi = *(const int4*)&lidx[buf][tid * 4];
      int4 vi = *(const int4*)&lval[buf][tid * 4];
      bi.x = vi.x ? bi.x : -1;
      bi.y = vi.y ? bi.y : -1;
      bi.z = vi.z ? bi.z : -1;
      bi.w = vi.w ? bi.w : -1;
      *(int4*)&lidx[buf][tid * 4] = bi;
    }
    __syncthreads();                           // packed indices visible
    if (t + 1 < TILES_PER_BLOCK) {             // overlap next map fetch with math
      const int p1 = p0 + TILE;
      async_copy_b128(imap + p1 + tid * 4, &lidx[buf ^ 1][tid * 4]);
      async_copy_b128(vmap + p1 + tid * 4, &lval[buf ^ 1][tid * 4]);
    }
#else
    __syncthreads();                           // done reading this buffer (prev round)
    {
      int4 bi = *(const int4*)(imap + p0 + tid * 4);
      int4 vi = *(const int4*)(vmap + p0 + tid * 4);
      bi.x = vi.x ? bi.x : -1;
      bi.y = vi.y ? bi.y : -1;
      bi.z = vi.z ? bi.z : -1;
      bi.w = vi.w ? bi.w : -1;
      *(int4*)&lidx[buf][tid * 4] = bi;
    }
    __syncthreads();
#endif

    // Stream 16 channels x 1024 pixels: NT global_load_b128 + ds_add_f32 scatter.
    const int i0 = tid * 4;
    const int bin0 = lidx[buf][i0 + 0];
    const int bin1 = lidx[buf][i0 + 1];
    const int bin2 = lidx[buf][i0 + 2];
    const int bin3 = lidx[buf][i0 + 3];
#pragma unroll 4
    for (int c = 0; c < CG; ++c) {
      // Single-use stream: non-temporal so 268 MB doesn't evict hot L2 lines.
      const v4f x = __builtin_nontemporal_load(
          (const v4f*)(in_base + (size_t)c * HW_ + p0 + i0));
      float* accc = acc + c * L_;
      if (bin0 >= 0) lds_fadd(accc + bin0, x.x);
      if (bin1 >= 0) lds_fadd(accc + bin1, x.y);
      if (bin2 >= 0) lds_fadd(accc + bin2, x.z);
      if (bin3 >= 0) lds_fadd(accc + bin3, x.w);
    }
    buf ^= 1;
  }

  __syncthreads();
  // Flush private bins: 8192 global f32 atomics per block (vs 131072 unprivatized).
  float* dst = out + ((size_t)b * C_ + (size_t)cg * CG) * L_;
  for (int j = tid; j < CG * L_; j += THREADS) {
    const float v = acc[j];
    if (v != 0.0f) glb_fadd(dst + j, v);
  }
}

__global__ __launch_bounds__(THREADS) void vlp_finalize_kernel(
    float* __restrict__ out, const int* __restrict__ count) {
  const int g = blockIdx.x * THREADS + threadIdx.x;   // < B*C*L
  const int l = g & (L_ - 1);
  const int b = g / (C_ * L_);
  out[g] = out[g] / (float)count[b * L_ + l];
}

extern "C" void kernel_launch(void* const* d_in, const int* in_sizes, int n_in,
                              void* d_out, int out_size, void* d_ws, size_t ws_size,
                              hipStream_t stream) {
  (void)in_sizes; (void)n_in; (void)d_ws; (void)ws_size; (void)out_size;
  const float* input  = (const float*)d_in[0];
  const int*   indmap = (const int*)d_in[1];
  const int*   count  = (const int*)d_in[2];
  const int*   valid  = (const int*)d_in[3];
  float*       out    = (float*)d_out;

  const int n = B_ * C_ * L_;  // 131072

  vlp_zero_kernel<<<n / THREADS, THREADS, 0, stream>>>(out, n);

  dim3 grid(HW_ / (TILE * TILES_PER_BLOCK), B_, C_ / CG);  // (32, 8, 2)
  vlp_pool_kernel<<<grid, THREADS, 0, stream>>>(input, indmap, valid, out);

  vlp_finalize_kernel<<<n / THREADS, THREADS, 0, stream>>>(out, count);
}